// SCSTCriterion_57990648430747
// MI455X (gfx1250) — compile-verified
//
#include <hip/hip_runtime.h>

// SCST criterion for MI455X (gfx1250, wave32).
//
// Roofline: total useful traffic ~40KB (4864 gathered f32 logprobs + 4608 i32
// targets) -> ~2ns at 23.3 TB/s. Latency/launch bound, so: ONE kernel, ONE
// workgroup (8 wave32s), gathers overlapped across waves, reduction done
// in-register with V_WMMA_F32_16X16X4_F32 (full f32 precision, deterministic
// order -> safe under graph replay; float global atomics would not be).

typedef float v2f __attribute__((ext_vector_type(2)));
typedef float v8f __attribute__((ext_vector_type(8)));

#define B_SZ 256
#define L_SZ 20
#define V_SZ 9488
#define S_SZ 18
#define TLEN 19   // L-1

__device__ __forceinline__ float rlane_f32(float v, int l) {
  return __int_as_float(__builtin_amdgcn_readlane(__float_as_int(v), l));
}

// One-instruction cross-half-wave reduction in full f32:
// A[m,0..3] = {p[m], 0, p[m+16], 0}, B = ones  =>  D[m,n] = p[m] + p[m+16].
// Column 0 of D lives in lane 0 (M=0..7 in d[0..7]) and lane 16 (M=8..15).
__device__ __forceinline__ float wave_reduce_wmma_f32(float p) {
  v2f a; a.x = p;    a.y = 0.0f;
  v2f b; b.x = 1.0f; b.y = 1.0f;          // all-ones B: layout-independent
  v8f c = {};
  c = __builtin_amdgcn_wmma_f32_16x16x4_f32(
      /*neg_a=*/false, a, /*neg_b=*/false, b,
      /*c_mod=*/(short)0, c, /*reuse_a=*/false, /*reuse_b=*/false);
  float s = ((c[0] + c[1]) + (c[2] + c[3])) + ((c[4] + c[5]) + (c[6] + c[7]));
  return rlane_f32(s, 0) + rlane_f32(s, 16);   // M=0..7 half + M=8..15 half
}

__global__ __launch_bounds__(256)
void scst_criterion_kernel(const float* __restrict__ inp,    // [B,L,V]
                           const int*   __restrict__ target, // [B,S]
                           const float* __restrict__ reward,
                           const float* __restrict__ reward1,
                           float* __restrict__ out) {
  __shared__ float lds_acc[256];
  __shared__ float lds_cnt[256];

  const int tid  = threadIdx.x;
  const int wave = tid >> 5;
  const int lane = tid & 31;

  float acc = 0.0f;
  float cnt = 0.0f;

  // 8 waves x 32 iterations = 256 rows; one row per wave-iteration,
  // lane t handles timestep t (lane 18 is the implicit pad zero).
  for (int i = 0; i < 32; ++i) {
    const int b = wave + (i << 3);
    int tg = (lane < S_SZ) ? target[b * S_SZ + lane] : 0;

    // mask[t] = 1 iff t < position of SECOND zero in row (pad counts).
    unsigned zball  = __builtin_amdgcn_ballot_w32(tg == 0);
    unsigned zmask  = zball & 0x7FFFFu;        // restrict to lanes 0..18
    unsigned zmask2 = zmask & (zmask - 1u);    // clear first zero
    int z2 = zmask2 ? __builtin_ctz(zmask2) : TLEN;

    if (lane < z2 && lane < TLEN) {
      size_t idx = ((size_t)b * L_SZ + (size_t)(lane + 1)) * (size_t)V_SZ
                 + (size_t)tg;
      acc += inp[idx];
      cnt += 1.0f;
    }
  }

  lds_acc[tid] = acc;
  lds_cnt[tid] = cnt;
  __syncthreads();

  // Every wave redundantly reduces all 256 partials (keeps the code path
  // uniform so EXEC is all-ones when the WMMAs issue).
  float pa = 0.0f, pc = 0.0f;
  for (int j = 0; j < 8; ++j) {
    pa += lds_acc[lane + (j << 5)];
    pc += lds_cnt[lane + (j << 5)];
  }

  float tot_sum = wave_reduce_wmma_f32(pa);
  float tot_cnt = wave_reduce_wmma_f32(pc);   // exact: small integers in f32

  if (tid == 0) {
    float rd = fmaxf(reward[0] - reward1[0], 1.0f);
    out[0] = -(tot_sum / tot_cnt) * rd;
  }
}

extern "C" void kernel_launch(void* const* d_in, const int* in_sizes, int n_in,
                              void* d_out, int out_size, void* d_ws, size_t ws_size,
                              hipStream_t stream) {
  (void)in_sizes; (void)n_in; (void)d_ws; (void)ws_size; (void)out_size;
  const float* inp     = (const float*)d_in[0];
  const int*   target  = (const int*)  d_in[1];
  const float* reward  = (const float*)d_in[2];
  const float* reward1 = (const float*)d_in[3];
  float* out = (float*)d_out;
  scst_criterion_kernel<<<dim3(1), dim3(256), 0, stream>>>(
      inp, target, reward, reward1, out);
}